// One_hot_CNN_5703716569443
// MI455X (gfx1250) — compile-verified
//
#include <hip/hip_runtime.h>

#define OC     128
#define V      30000
#define KW     5
#define L_TOT  2560          // 32 x 40 x 64 tokens -> (32, 2560)
#define LOUT   (L_TOT - KW + 1)   // 2556
#define NB     32
#define TL     64            // l-positions per block
#define TC_OUT 4

typedef __attribute__((ext_vector_type(2))) float v2f;
typedef __attribute__((ext_vector_type(8))) float v8f;

__global__ void zero_pooled_kernel(float* __restrict__ pooled) {
    int i = blockIdx.x * blockDim.x + threadIdx.x;
    if (i < NB * OC) pooled[i] = 0.0f;
}

// One block per (l-tile, batch). threadIdx.x == output channel.
// Sliding-window: token at position p contributes K[oc, id_p, w] to conv[p-w].
__global__ __launch_bounds__(OC) void conv_gather_maxpool_kernel(
    const int*   __restrict__ tokens,   // (NB, L_TOT)
    const float* __restrict__ Ktab,     // (OC, V, KW) contiguous
    float*       __restrict__ pooled)   // (NB, OC)
{
    __shared__ int ids[TL + KW - 1];

    const int b  = blockIdx.y;
    const int l0 = blockIdx.x * TL;
    const int oc = threadIdx.x;

    if (threadIdx.x < TL + KW - 1) {
        int p = l0 + threadIdx.x;
        ids[threadIdx.x] = (p < L_TOT) ? tokens[b * L_TOT + p] : 0;
    }
    __syncthreads();

    const float* Koc = Ktab + (size_t)oc * (size_t)V * KW;
    const int lend = min(l0 + TL, LOUT);

    float a0 = 0.f, a1 = 0.f, a2 = 0.f, a3 = 0.f, a4 = 0.f;
    float m = 0.0f;   // ReLU folded in: max starts at 0

    #pragma unroll 1
    for (int j = 0; j < TL + KW - 1; ++j) {
        const int p = l0 + j;
        if (p >= L_TOT) break;

        const float* kp = Koc + (size_t)ids[j] * KW;   // 20 contiguous bytes
        if (j + 1 < TL + KW - 1) {
            __builtin_prefetch(Koc + (size_t)ids[j + 1] * KW, 0, 3);
        }
        const float k0 = kp[0], k1 = kp[1], k2 = kp[2], k3 = kp[3], k4 = kp[4];
        a0 += k0; a1 += k1; a2 += k2; a3 += k3; a4 += k4;

        // a4 now holds the completed conv value for l = p - (KW-1)
        const int l = p - (KW - 1);
        if (l >= l0 && l < lend) m = fmaxf(m, a4);

        a4 = a3; a3 = a2; a2 = a1; a1 = a0; a0 = 0.f;
    }

    // m >= 0, so IEEE bit pattern order == integer order
    atomicMax((int*)&pooled[b * OC + oc], __float_as_int(m));
}

// out(32x4) = pooled(32x128) @ fc1_w^T(128x4) + fc1_b, via V_WMMA_F32_16X16X4_F32.
// Single wave32 block. Two 16-row M tiles, K accumulated in steps of 4.
__global__ __launch_bounds__(32) void fc_wmma_kernel(
    const float* __restrict__ pooled,  // (32, 128)
    const float* __restrict__ fc1_w,   // (4, 128)
    const float* __restrict__ fc1_b,   // (4,)
    float*       __restrict__ out)     // (32, 4)
{
    const int lane = threadIdx.x;
    const int half = lane >> 4;   // lane-half selects K columns {0,1} vs {2,3}
    const int idx  = lane & 15;   // A: row m ; B/D: column n

    v8f acc0 = {};   // rows 0..15
    v8f acc1 = {};   // rows 16..31

    for (int k0 = 0; k0 < OC; k0 += 4) {
        const int ka = k0 + half * 2;

        v2f a0, a1, bm;
        // A 16x4 layout: lanes 0-15 vgpr{0,1} = K cols {0,1}; lanes 16-31 = cols {2,3}
        a0.x = pooled[idx * OC + ka];
        a0.y = pooled[idx * OC + ka + 1];
        a1.x = pooled[(idx + 16) * OC + ka];
        a1.y = pooled[(idx + 16) * OC + ka + 1];
        // B 4x16 layout: vgpr j holds row k0+half*2+j at column n = idx; pad n>=4 with 0
        if (idx < TC_OUT) {
            bm.x = fc1_w[idx * OC + ka];
            bm.y = fc1_w[idx * OC + ka + 1];
        } else {
            bm.x = 0.f;
            bm.y = 0.f;
        }

        acc0 = __builtin_amdgcn_wmma_f32_16x16x4_f32(
            false, a0, false, bm, (short)0, acc0, false, false);
        acc1 = __builtin_amdgcn_wmma_f32_16x16x4_f32(
            false, a1, false, bm, (short)0, acc1, false, false);
    }

    // D layout: lane-half -> m high bit (m = half*8 + r), n = idx
    if (idx < TC_OUT) {
        const float bias = fc1_b[idx];
        #pragma unroll
        for (int r = 0; r < 8; ++r) {
            const int m0 = half * 8 + r;
            out[m0 * TC_OUT + idx]        = acc0[r] + bias;
            out[(16 + m0) * TC_OUT + idx] = acc1[r] + bias;
        }
    }
}

extern "C" void kernel_launch(void* const* d_in, const int* in_sizes, int n_in,
                              void* d_out, int out_size, void* d_ws, size_t ws_size,
                              hipStream_t stream) {
    (void)in_sizes; (void)n_in; (void)out_size; (void)ws_size;

    const int*   tokens  = (const int*)d_in[0];
    const float* kernel1 = (const float*)d_in[1];   // (OC, V, 1, KW)
    const float* fc1_w   = (const float*)d_in[2];   // (4, OC)
    const float* fc1_b   = (const float*)d_in[3];   // (4,)
    float*       out     = (float*)d_out;
    float*       pooled  = (float*)d_ws;            // (NB, OC) = 16 KB scratch

    zero_pooled_kernel<<<dim3((NB * OC + 255) / 256), dim3(256), 0, stream>>>(pooled);

    dim3 grid((LOUT + TL - 1) / TL, NB);            // 40 x 32 blocks, 128 thr each
    conv_gather_maxpool_kernel<<<grid, dim3(OC), 0, stream>>>(tokens, kernel1, pooled);

    fc_wmma_kernel<<<dim3(1), dim3(32), 0, stream>>>(pooled, fc1_w, fc1_b, out);
}